// Perspective_Layer_44487271252510
// MI455X (gfx1250) — compile-verified
//
#include <hip/hip_runtime.h>

// Perspective warp layer for MI455X (gfx1250, wave32).
// Memory-bound: ~256MB HBM traffic (input stays L2-resident, output streamed
// with non-temporal stores). The reference einsum('tcij,jp') is a K=3 GEMM,
// computed here with V_WMMA_F32_16X16X4_F32 (K padded to 4).

typedef float v2f __attribute__((ext_vector_type(2)));
typedef float v8f __attribute__((ext_vector_type(8)));

#define N_  8
#define C_  32
#define H_  224
#define W_  224
#define TM_ 4

static __device__ __forceinline__ int iclamp(int v, int lo, int hi) {
    return v < lo ? lo : (v > hi ? hi : v);
}

__global__ __launch_bounds__(256)
void persp_wmma_kernel(const float* __restrict__ in,   // (N,C,H,W)
                       const float* __restrict__ wt,   // (TM,C,8)
                       float* __restrict__ out) {      // (N,C*TM,H,W)
    const int A    = blockIdx.x;   // output row  -> drives x-chain (xs[A])
    const int c    = blockIdx.y;
    const int t    = blockIdx.z;
    const int tid  = threadIdx.x;  // output col B
    const int lane = tid & 31;
    const int wave = tid >> 5;

    // Per-(t,c) 3x3 matrix rows (m22 == 1). Block-uniform -> scalar loads.
    const float* m = wt + ((size_t)t * C_ + c) * 8;
    const float m00 = m[0], m01 = m[1], m02 = m[2];
    const float m10 = m[3], m11 = m[4], m12 = m[5];
    const float m20 = m[6], m21 = m[7];

    __shared__ float sX[256], sY[256], sO[256];

    const float step  = 2.0f / 223.0f;
    const float x_lin = -1.0f + step * (float)A;   // uniform across block

    // ---- A-matrix for V_WMMA_F32_16X16X4_F32 (16x4, rows 0..2 live) ----
    // A(M,K): K=0 -> VGPR0 lane M ; K=1 -> VGPR1 lane M ;
    //         K=2 -> VGPR0 lane 16+M ; K=3 -> VGPR1 lane 16+M (pad = 0)
    float a0 = 0.0f, a1 = 0.0f;
    if      (lane == 0)  { a0 = m00;  a1 = m01; }
    else if (lane == 1)  { a0 = m10;  a1 = m11; }
    else if (lane == 2)  { a0 = m20;  a1 = m21; }
    else if (lane == 16) { a0 = m02; }
    else if (lane == 17) { a0 = m12; }
    else if (lane == 18) { a0 = 1.0f; }           // m22
    v2f Amat; Amat[0] = a0; Amat[1] = a1;

    // Each wave computes [X;Y;O] for its 32 pixels via two 16-column WMMAs.
#pragma unroll
    for (int g = 0; g < 2; ++g) {
        const int colI   = wave * 32 + g * 16 + (lane & 15);
        const float y_ln = -1.0f + step * (float)colI;
        // B(K,N): K=0 row (x_lin) lanes 0-15 / K=2 row (ones) lanes 16-31 in VGPR0
        //         K=1 row (y_lin) lanes 0-15 / K=3 row (pad)  lanes 16-31 in VGPR1
        v2f Bmat;
        Bmat[0] = (lane < 16) ? x_lin : 1.0f;
        Bmat[1] = (lane < 16) ? y_ln  : 0.0f;
        v8f Cacc = {};
        v8f D = __builtin_amdgcn_wmma_f32_16x16x4_f32(
            /*neg_a=*/false, Amat, /*neg_b=*/false, Bmat,
            /*c_mod=*/(short)0, Cacc, /*reuse_a=*/false, /*reuse_b=*/false);
        // D row r, col n -> VGPR r, lane n (lanes 0-15). Rows 0,1,2 = X,Y,O.
        if (lane < 16) {
            sX[colI] = D[0];
            sY[colI] = D[1];
            sO[colI] = D[2];
        }
    }
    __syncthreads();

    if (tid >= W_) return;
    const int B = tid;

    // ---- per-pixel sampling coords (computed once, reused over N=8) ----
    const float X = sX[B], Y = sY[B], O = sO[B];
    const float x_s = X / O;
    const float y_s = Y / O;
    const float x = 0.5f * ((x_s + 1.0f) * 222.0f);   // (max_x - 1) = 222
    const float y = 0.5f * ((y_s + 1.0f) * 222.0f);

    int ix0 = (int)floorf(x);
    int iy0 = (int)floorf(y);
    int ix1 = ix0 + 1;
    int iy1 = iy0 + 1;
    ix0 = iclamp(ix0, 0, W_ - 1);  ix1 = iclamp(ix1, 0, W_ - 1);
    iy0 = iclamp(iy0, 0, H_ - 1);  iy1 = iclamp(iy1, 0, H_ - 1);

    const float x0f = (float)ix0, x1f = (float)ix1;
    const float y0f = (float)iy0, y1f = (float)iy1;
    // Reference's (nonstandard) weights — replicated exactly.
    const float wa = (x1f - x) * (y1f - y);
    const float wb = (x1f - x) * (y1f - y0f);
    const float wc = (x - x0f) * (y1f - y);
    const float wd = (x - x0f) * (y - y0f);

    const size_t off00 = (size_t)iy0 * W_ + ix0;
    const size_t off10 = (size_t)iy1 * W_ + ix0;
    const size_t off01 = (size_t)iy0 * W_ + ix1;
    const size_t off11 = (size_t)iy1 * W_ + ix1;

    const size_t plane  = (size_t)H_ * W_;
    const size_t in_c   = (size_t)c * plane;
    const size_t out_ct = (size_t)(c * TM_ + t);

#pragma unroll
    for (int n = 0; n < N_; ++n) {
        const float* p = in + (size_t)n * C_ * plane + in_c;   // L2-resident
        const float Ia = p[off00];
        const float Ib = p[off10];
        const float Ic = p[off01];
        const float Id = p[off11];
        const float v = wa * Ia + wb * Ib + wc * Ic + wd * Id;
        float* po = out + (((size_t)n * (C_ * TM_) + out_ct) * H_ + A) * W_ + B;
        __builtin_nontemporal_store(v, po);   // stream 205MB out, keep L2 for input
    }
}

extern "C" void kernel_launch(void* const* d_in, const int* in_sizes, int n_in,
                              void* d_out, int out_size, void* d_ws, size_t ws_size,
                              hipStream_t stream) {
    const float* inputs = (const float*)d_in[0];   // (8,32,224,224) f32
    const float* wt     = (const float*)d_in[1];   // (4,32,8) f32
    float* out          = (float*)d_out;           // (8,128,224,224) f32

    dim3 grid(H_, C_, TM_);   // one block per (output-row A, channel c, transform t)
    persp_wmma_kernel<<<grid, 256, 0, stream>>>(inputs, wt, out);
}